// KalmanFilter_68341519614192
// MI455X (gfx1250) — compile-verified
//
// MI455X (gfx1250) Kalman filter scan: one wave32 per batch chain, one wave
// per workgroup (barriers are HW NOPs for single-wave workgroups but keep
// compiler-side LDS ordering). Covariance P lives in registers in the WMMA
// C-layout (8 VGPRs x 32 lanes); all 16x16 products use
// V_WMMA_F32_16X16X4_F32 (4 K-chunks, 24 WMMA per step). Cross-half data
// movement uses v_permlanex16 (VALU) instead of ds_bpermute; Gauss-Jordan
// pivot rows broadcast via v_readlane into SGPRs.
// Roofline: outputs ~563 MB dominate (HBM floor ~24us @ 23.3 TB/s);
// compute ~16 GFLOP total, far under any WMMA ceiling.
#include <hip/hip_runtime.h>

typedef float v2f __attribute__((ext_vector_type(2)));
typedef float v8f __attribute__((ext_vector_type(8)));

#define NSd 16
#define MSd 8
#define Td  200
#define BLOCK_T 32          // one wave32 per workgroup

// value of lane (lane ^ 16): cross-half fetch with identity nibble selectors
__device__ __forceinline__ float permx16(float v) {
  return __int_as_float(__builtin_amdgcn_permlanex16(
      __float_as_int(v), __float_as_int(v), 0x76543210, (int)0xfedcba98,
      false, false));
}

// uniform broadcast from a compile-time lane (v_readlane -> SGPR)
__device__ __forceinline__ float bcast_lane(float v, int l) {
  return __int_as_float(__builtin_amdgcn_readlane(__float_as_int(v), l));
}

// D = A*B + C, 16x16x4 fp32 WMMA.
// (neg_a, A, neg_b, B, c_mod, C, reuse_a, reuse_b)
__device__ __forceinline__ v8f wmma4(v2f a, v2f b, v8f c) {
  return __builtin_amdgcn_wmma_f32_16x16x4_f32(false, a, false, b, (short)0, c,
                                               false, false);
}

// K-chunk (k0 in {0,4,8,12}) fragment of a 16x16 matrix M held in C/D layout.
//   frag.x = M[k0 + (hi?2:0)][lane&15], frag.y = M[k0 + (hi?3:1)][lane&15]
// Used as a B operand it represents M; used as an A operand it represents M^T
// (== M itself when M is symmetric). 2 permlanex16 + 2 cndmask per chunk.
__device__ __forceinline__ v2f bfragC(v8f m, int k0, bool hi) {
  int b = k0 & 7;
  v2f r;
  if (k0 & 8) {
    // rows >= 8 live in the high half: lo lanes fetch across, hi lanes keep own
    float tx = permx16(m[b + 0]);
    float ty = permx16(m[b + 1]);
    r.x = hi ? m[b + 2] : tx;
    r.y = hi ? m[b + 3] : ty;
  } else {
    // rows < 8 live in the low half: hi lanes fetch across, lo lanes keep own
    float tx = permx16(m[b + 2]);
    float ty = permx16(m[b + 3]);
    r.x = hi ? tx : m[b + 0];
    r.y = hi ? ty : m[b + 1];
  }
  return r;
}

__global__ __launch_bounds__(BLOCK_T)
void kalman_scan_kernel(const float* __restrict__ in_y,     // (B,T,8)
                        const float* __restrict__ m0g,      // (16)
                        const float* __restrict__ logd,     // (16)
                        const float* __restrict__ offd,     // (120)
                        const float* __restrict__ Fg,       // (16,16)
                        const float* __restrict__ Hg,       // (8,16)
                        const float* __restrict__ Qg,       // (16,16)
                        const float* __restrict__ Rg,       // (8,8)
                        float* __restrict__ out_sm,         // (B,T,16)
                        float* __restrict__ out_P,          // (B,T,16,16)
                        float* __restrict__ out_mm,         // (B,T,8)
                        float* __restrict__ out_mc)         // (B,T,8,8)
{
  __shared__ float Fs[256];
  __shared__ float Hs[128];
  __shared__ float sLt[256];
  __shared__ float sP0[256];
  __shared__ float sS[64];
  __shared__ float sSinv[64];
  __shared__ float sKT[128];
  __shared__ float sresid[8];
  __shared__ float smv[16];
  __shared__ float smtmp[16];

  const int  lane  = threadIdx.x & 31;
  const int  batch = blockIdx.x;
  const int  col   = lane & 15;
  const bool hi    = lane >= 16;

  // ---------------- one-time constants (per workgroup) ----------------
  for (int idx = lane; idx < 256; idx += 32) {
    Fs[idx] = Fg[idx];
    int i = idx >> 4, j = idx & 15;
    float l = 0.f;
    if (i == j)      l = expf(logd[i]);
    else if (j < i)  l = offd[(i * (i - 1)) / 2 + j];
    sLt[idx] = l;
  }
  for (int idx = lane; idx < 128; idx += 32) Hs[idx] = Hg[idx];
  __syncthreads();
  for (int idx = lane; idx < 256; idx += 32) {   // P0 = L L^T
    int i = idx >> 4, j = idx & 15;
    float acc = 0.f;
#pragma unroll
    for (int k = 0; k < 16; ++k) acc += sLt[i * 16 + k] * sLt[j * 16 + k];
    sP0[idx] = acc;
  }
  if (lane < 16) smv[lane] = m0g[lane];
  __syncthreads();

  // Constant fragments. FA[c] serves both "F as A-operand" and "F^T as
  // B-operand" (identical register contents); same for HA[c] with Hpad.
  v2f FA[4], HA[4];
#pragma unroll
  for (int c = 0; c < 4; ++c) {
    int k = 4 * c + (hi ? 2 : 0);
    FA[c].x = Fg[col * 16 + k];
    FA[c].y = Fg[col * 16 + k + 1];
    HA[c].x = (col < 8) ? Hg[col * 16 + k]     : 0.f;
    HA[c].y = (col < 8) ? Hg[col * 16 + k + 1] : 0.f;
  }
  v8f Qc, Rc;
#pragma unroll
  for (int r = 0; r < 8; ++r) {
    int row = r + (hi ? 8 : 0);
    Qc[r] = Qg[row * 16 + col];
    Rc[r] = (!hi && col < 8) ? Rg[r * 8 + col] : 0.f;   // Rpad C-layout
  }

  v8f P;                                                // P0 in C-layout
#pragma unroll
  for (int r = 0; r < 8; ++r) P[r] = sP0[(r + (hi ? 8 : 0)) * 16 + col];

  const size_t bT = (size_t)batch * Td;

  for (int t = 0; t < Td; ++t) {
    const size_t bt = bT + t;

    // ---- emit state mean / state cov at t ----
    if (lane < 16) out_sm[bt * 16 + lane] = smv[lane];
#pragma unroll
    for (int r = 0; r < 8; ++r)
      out_P[bt * 256 + (size_t)((r + (hi ? 8 : 0)) * 16 + col)] = P[r];

    // ---- PHt = P * Hpad^T  (A = P by symmetry; B = Hpad^T == HA) ----
    v2f Pb[4];
#pragma unroll
    for (int c = 0; c < 4; ++c) Pb[c] = bfragC(P, 4 * c, hi);
    v8f PHt = {};
#pragma unroll
    for (int c = 0; c < 4; ++c) PHt = wmma4(Pb[c], HA[c], PHt);

    // ---- S = Hpad * PHt + Rpad ----
    v2f PHb[4];
#pragma unroll
    for (int c = 0; c < 4; ++c) PHb[c] = bfragC(PHt, 4 * c, hi);
    v8f S = Rc;
#pragma unroll
    for (int c = 0; c < 4; ++c) S = wmma4(HA[c], PHb[c], S);

    // ---- mu = H m; emit measurement mean / cov at t ----
    float mu = 0.f;
    if (lane < 8) {
#pragma unroll
      for (int k = 0; k < 16; ++k) mu += Hs[lane * 16 + k] * smv[k];
      out_mm[bt * 8 + lane] = mu;
#pragma unroll
      for (int r = 0; r < 8; ++r) out_mc[bt * 64 + r * 8 + lane] = S[r];
    }

    if (t == Td - 1) break;   // last step emits only (uniform)

    // ---- stage S rows + residual into LDS ----
    if (lane < 8) {
      float y = in_y[bt * 8 + lane];
      sresid[lane] = y - mu;
#pragma unroll
      for (int r = 0; r < 8; ++r) sS[r * 8 + lane] = S[r];
    }
    __syncthreads();

    // ---- register Gauss-Jordan inverse of S (8x8 SPD), rows per lane ----
    {
      int r = lane & 7;
      float a[16];
#pragma unroll
      for (int j = 0; j < 8; ++j) a[j] = sS[r * 8 + j];
#pragma unroll
      for (int j = 0; j < 8; ++j) a[8 + j] = (j == r) ? 1.f : 0.f;
#pragma unroll
      for (int p = 0; p < 8; ++p) {
        float pr[16];
#pragma unroll
        for (int j = 0; j < 16; ++j) pr[j] = bcast_lane(a[j], p);
        float invp = 1.0f / pr[p];
        if (r == p) {
#pragma unroll
          for (int j = 0; j < 16; ++j) a[j] = pr[j] * invp;
        } else {
          float f = a[p];
#pragma unroll
          for (int j = 0; j < 16; ++j) a[j] -= f * (pr[j] * invp);
        }
      }
      if (lane < 8) {
#pragma unroll
        for (int j = 0; j < 8; ++j) sSinv[r * 8 + j] = a[8 + j];
      }
    }
    __syncthreads();

    // ---- HP = Hpad * P ----
    v8f HP = {};
#pragma unroll
    for (int c = 0; c < 4; ++c) HP = wmma4(HA[c], Pb[c], HP);
    v2f HPb[2];
    HPb[0] = bfragC(HP, 0, hi);
    HPb[1] = bfragC(HP, 4, hi);

    // ---- KT = Sinv_pad * HP  (K^T = Sinv * HP since Sinv symmetric) ----
    v2f SA[2];
#pragma unroll
    for (int c = 0; c < 2; ++c) {
      int k = 4 * c + (hi ? 2 : 0);
      bool ok = col < 8;
      SA[c].x = ok ? sSinv[col * 8 + k]     : 0.f;
      SA[c].y = ok ? sSinv[col * 8 + k + 1] : 0.f;
    }
    v8f KT = {};
    KT = wmma4(SA[0], HPb[0], KT);
    KT = wmma4(SA[1], HPb[1], KT);

    // ---- X = K * HP   (A = use-as-A(KT) == K);  P_u = P - X ----
    v2f Ka[2];
    Ka[0] = bfragC(KT, 0, hi);
    Ka[1] = bfragC(KT, 4, hi);
    v8f X = {};
    X = wmma4(Ka[0], HPb[0], X);
    X = wmma4(Ka[1], HPb[1], X);
    v8f Pu = P - X;

    // ---- mean update: m_u = m + K resid; m_p = F m_u ----
    if (lane < 16) {
#pragma unroll
      for (int r = 0; r < 8; ++r) sKT[r * 16 + lane] = KT[r];
    }
    __syncthreads();
    if (lane < 16) {
      float mui = smv[lane];
#pragma unroll
      for (int j = 0; j < 8; ++j) mui += sKT[j * 16 + lane] * sresid[j];
      smtmp[lane] = mui;
    }
    __syncthreads();
    float mp = 0.f;
    if (lane < 16) {
#pragma unroll
      for (int j = 0; j < 16; ++j) mp += Fs[lane * 16 + j] * smtmp[j];
    }

    // ---- predict: G = P_u * F^T ; P = F * G + Q ----
    v2f Pub[4];
#pragma unroll
    for (int c = 0; c < 4; ++c) Pub[c] = bfragC(Pu, 4 * c, hi);
    v8f G = {};
#pragma unroll
    for (int c = 0; c < 4; ++c) G = wmma4(Pub[c], FA[c], G);
    v2f Gb[4];
#pragma unroll
    for (int c = 0; c < 4; ++c) Gb[c] = bfragC(G, 4 * c, hi);
    v8f Pn = Qc;
#pragma unroll
    for (int c = 0; c < 4; ++c) Pn = wmma4(FA[c], Gb[c], Pn);
    P = Pn;

    if (lane < 16) smv[lane] = mp;
    __syncthreads();
  }
}

extern "C" void kernel_launch(void* const* d_in, const int* in_sizes, int n_in,
                              void* d_out, int out_size, void* d_ws,
                              size_t ws_size, hipStream_t stream) {
  (void)n_in; (void)out_size; (void)d_ws; (void)ws_size;
  const float* in_y = (const float*)d_in[0];
  const float* m0g  = (const float*)d_in[1];
  const float* logd = (const float*)d_in[2];
  const float* offd = (const float*)d_in[3];
  const float* Fg   = (const float*)d_in[4];
  const float* Hg   = (const float*)d_in[5];
  const float* Qg   = (const float*)d_in[6];
  const float* Rg   = (const float*)d_in[7];

  int Bc = in_sizes[0] / (Td * MSd);          // 2048
  size_t BT = (size_t)Bc * Td;
  float* out    = (float*)d_out;
  float* out_sm = out;
  float* out_P  = out_sm + BT * NSd;
  float* out_mm = out_P + BT * NSd * NSd;
  float* out_mc = out_mm + BT * MSd;

  kalman_scan_kernel<<<Bc, BLOCK_T, 0, stream>>>(
      in_y, m0g, logd, offd, Fg, Hg, Qg, Rg,
      out_sm, out_P, out_mm, out_mc);
}